// Sparsemax_79620103733677
// MI455X (gfx1250) — compile-verified
//
#include <hip/hip_runtime.h>

// Sparsemax over rows: x [16384, 4096] f32 -> out [16384, 4096] f32.
// One workgroup (256 threads, 8 wave32) per row. Row staged to LDS via the
// CDNA5 async global->LDS path (ASYNCcnt-tracked), held in registers
// (16 f32/thread), tau found exactly with Michelot's fixed-point iteration.

constexpr int D   = 4096;
constexpr int TPB = 256;
constexpr int F4T = 4;            // float4 chunks per thread (16 floats)
constexpr int NWAVES = TPB / 32;  // 8 wave32

// Pointer types matching the async-to-LDS builtin signature:
// (AS1 int4*, AS3 int4*, imm offset, imm cpol)
typedef int v4i __attribute__((ext_vector_type(4)));
typedef __attribute__((address_space(1))) v4i* as1_v4i_ptr;
typedef __attribute__((address_space(3))) v4i* as3_v4i_ptr;

__device__ __forceinline__ void blockReduce2(float& a, float& b,
                                             float (*red)[NWAVES][2],
                                             int buf, int lane, int wid) {
  // wave32 butterfly reduction of (a, b)
#pragma unroll
  for (int m = 16; m >= 1; m >>= 1) {
    a += __shfl_xor(a, m, 32);
    b += __shfl_xor(b, m, 32);
  }
  if (lane == 0) { red[buf][wid][0] = a; red[buf][wid][1] = b; }
  __syncthreads();
  float sa = 0.0f, sb = 0.0f;
#pragma unroll
  for (int w = 0; w < NWAVES; ++w) { sa += red[buf][w][0]; sb += red[buf][w][1]; }
  a = sa; b = sb;   // identical value & summation order in every lane
}

__global__ __launch_bounds__(TPB) void sparsemax_rows(
    const float* __restrict__ x, float* __restrict__ out) {
  __shared__ float row[D];                 // 16 KB staged row
  __shared__ float red[2][NWAVES][2];      // double-buffered reduction scratch

  const int tid  = threadIdx.x;
  const int lane = tid & 31;
  const int wid  = tid >> 5;
  const size_t base = (size_t)blockIdx.x * (size_t)D;
  const float4* gin  = (const float4*)(x + base);
  float4*       gout = (float4*)(out + base);
  float4*       lrow = (float4*)row;

  // ---- Stage row into LDS (async DMA path on gfx1250, ASYNCcnt-tracked) ----
#if __has_builtin(__builtin_amdgcn_global_load_async_to_lds_b128)
#pragma unroll
  for (int k = 0; k < F4T; ++k) {
    const int f4 = tid + k * TPB;
    __builtin_amdgcn_global_load_async_to_lds_b128(
        (as1_v4i_ptr)(void*)(gin + f4),
        (as3_v4i_ptr)(void*)(lrow + f4),
        0, 0);
  }
#if __has_builtin(__builtin_amdgcn_s_wait_asynccnt)
  __builtin_amdgcn_s_wait_asynccnt(0);
#else
  asm volatile("s_wait_asynccnt 0" ::: "memory");
#endif
#else
#pragma unroll
  for (int k = 0; k < F4T; ++k) {
    const int f4 = tid + k * TPB;
    lrow[f4] = gin[f4];
  }
#endif
  __syncthreads();

  // ---- Fan out 16 values/thread into registers ----
  float v[16];
#pragma unroll
  for (int k = 0; k < F4T; ++k) {
    float4 t = lrow[tid + k * TPB];
    v[4 * k + 0] = t.x; v[4 * k + 1] = t.y;
    v[4 * k + 2] = t.z; v[4 * k + 3] = t.w;
  }

  // ---- Michelot iteration: tau s.t. sum(max(x - tau, 0)) == 1 ----
  float s = 0.0f, c0 = 0.0f;
#pragma unroll
  for (int i = 0; i < 16; ++i) s += v[i];
  blockReduce2(s, c0, red, 0, lane, wid);
  float tau   = (s - 1.0f) * (1.0f / (float)D);
  float prevC = (float)D;

  for (int it = 0; it < 64; ++it) {
    float ls = 0.0f, lc = 0.0f;
#pragma unroll
    for (int i = 0; i < 16; ++i) {
      if (v[i] > tau) { ls += v[i]; lc += 1.0f; }
    }
    blockReduce2(ls, lc, red, (it + 1) & 1, lane, wid);
    tau = (ls - 1.0f) / lc;        // lc >= 1 always (max element stays active)
    if (lc == prevC) break;        // active set unchanged -> converged (uniform)
    prevC = lc;
  }

  // ---- Emit max(x - tau, 0) ----
#pragma unroll
  for (int k = 0; k < F4T; ++k) {
    float4 o;
    o.x = fmaxf(v[4 * k + 0] - tau, 0.0f);
    o.y = fmaxf(v[4 * k + 1] - tau, 0.0f);
    o.z = fmaxf(v[4 * k + 2] - tau, 0.0f);
    o.w = fmaxf(v[4 * k + 3] - tau, 0.0f);
    gout[tid + k * TPB] = o;
  }
}

extern "C" void kernel_launch(void* const* d_in, const int* in_sizes, int n_in,
                              void* d_out, int out_size, void* d_ws, size_t ws_size,
                              hipStream_t stream) {
  const float* x = (const float*)d_in[0];
  float* out     = (float*)d_out;
  const int rows = in_sizes[0] / D;   // 16384
  sparsemax_rows<<<dim3(rows), dim3(TPB), 0, stream>>>(x, out);
}